// SimpleGCNNet_20074677141980
// MI455X (gfx1250) — compile-verified
//
#include <hip/hip_runtime.h>
#include <hip/hip_bf16.h>

// ---------------------------------------------------------------------------
// SGConv (K=2) on gfx1250:
//   1) deg = 1 + segsum(clamp(ew), col);  dinv = deg>0 ? rsqrt(deg) : 0
//   2) norm[e] = dinv[row]*clamp(ew)*dinv[col]
//   3) hop: y = x*dinv^2 (self loop)  +  scatter_add(x[row]*norm -> col)
//      (edge metadata via wave-uniform scalar loads; payload via float4 +
//       global_atomic_add_f32)
//   4) out = y @ W + b  via V_WMMA_F32_16X16X4_F32, A-tile staged in LDS
// ---------------------------------------------------------------------------

#define NNODES 50000
#define NEDGES 1600000
#define TFEAT  128   // in features
#define HFEAT  128   // out features
#define EDGES_PER_BLOCK 8   // 256 threads = 8 wave32, one edge per wave

typedef __attribute__((ext_vector_type(2))) float v2f;
typedef __attribute__((ext_vector_type(8))) float v8f;

// ---------------- normalization ----------------

__global__ void init_deg_kernel(float* __restrict__ deg, int n) {
    int i = blockIdx.x * blockDim.x + threadIdx.x;
    if (i < n) deg[i] = 1.0f;                       // self-loop weight
}

__global__ void accum_deg_kernel(const float* __restrict__ ew,
                                 const int* __restrict__ col,
                                 float* __restrict__ deg, int e) {
    int i = blockIdx.x * blockDim.x + threadIdx.x;
    if (i >= e) return;
    float w = ew[i];
    w = fminf(fmaxf(w, -2.0f), 5.0f);               // straight-through clamp (fwd)
    unsafeAtomicAdd(&deg[col[i]], w);               // global_atomic_add_f32
}

__global__ void dinv_kernel(float* __restrict__ deg, int n) {
    int i = blockIdx.x * blockDim.x + threadIdx.x;
    if (i >= n) return;
    float d = deg[i];
    deg[i] = (d > 0.0f) ? rsqrtf(d) : 0.0f;         // in place: deg -> dinv
}

__global__ void norm_kernel(const float* __restrict__ ew,
                            const int* __restrict__ row,
                            const int* __restrict__ col,
                            const float* __restrict__ dinv,
                            float* __restrict__ norm, int e) {
    int i = blockIdx.x * blockDim.x + threadIdx.x;
    if (i >= e) return;
    float w = ew[i];
    w = fminf(fmaxf(w, -2.0f), 5.0f);
    norm[i] = dinv[row[i]] * w * dinv[col[i]];
}

// ---------------- propagation (one hop) ----------------

// y[i][:] = x[i][:] * dinv[i]^2   (self-loop term, also zero-initializes y)
__global__ void prop_init_kernel(const float* __restrict__ x,
                                 const float* __restrict__ dinv,
                                 float* __restrict__ y, int n) {
    int t = blockIdx.x * blockDim.x + threadIdx.x;  // one float4 per thread
    int node = t >> 5;                              // 32 float4 = 128 floats/node
    if (node >= n) return;
    float di = dinv[node];
    float s  = di * di;
    const float4* src = (const float4*)x + (size_t)t;
    float4*       dst = (float4*)y + (size_t)t;
    float4 v = *src;
    v.x *= s; v.y *= s; v.z *= s; v.w *= s;
    *dst = v;
}

// scatter: y[col] += x[row] * norm
// One wave32 per edge. Edge id is wave-uniform -> readfirstlane puts it in an
// SGPR so row/col/norm lower to scalar (SMEM) loads instead of 32 redundant
// vector lanes. Grid is exact (E % EDGES_PER_BLOCK == 0): no bounds guard,
// EXEC stays all-ones.
__global__ __launch_bounds__(256) void prop_edges_kernel(
        const float* __restrict__ x,
        const int* __restrict__ row,
        const int* __restrict__ col,
        const float* __restrict__ norm,
        float* __restrict__ y) {
    const int lane = threadIdx.x & 31;
    const int eid  = __builtin_amdgcn_readfirstlane(
                         blockIdx.x * EDGES_PER_BLOCK + (threadIdx.x >> 5));
    const float nrm = norm[eid];                    // uniform -> s_load_b32
    if (nrm == 0.0f) return;                        // wave-uniform skip
    const int r = row[eid];                         // uniform -> s_load_b32
    const int c = col[eid];
    const float4 v = ((const float4*)(x + (size_t)r * TFEAT))[lane];
    float* dst = y + (size_t)c * TFEAT + lane * 4;
    unsafeAtomicAdd(dst + 0, v.x * nrm);
    unsafeAtomicAdd(dst + 1, v.y * nrm);
    unsafeAtomicAdd(dst + 2, v.z * nrm);
    unsafeAtomicAdd(dst + 3, v.w * nrm);
}

// ---------------- final linear via fp32 WMMA ----------------
// Block = 256 threads = 8 waves. Block b -> rows [16b,16b+16), wave w -> cols
// [16w,16w+16). The 16x128 A tile is shared by all 8 waves -> stage it in LDS
// once (ds_store_b128), then each wave feeds V_WMMA_F32_16X16X4_F32 from LDS
// (ds_load_b64 pairs). 50000 = 3125*16 exactly -> no remainder, EXEC all-ones.
__global__ __launch_bounds__(256) void gemm_wmma_f32_kernel(
        const float* __restrict__ X,    // [NNODES, TFEAT]
        const float* __restrict__ W,    // [TFEAT, HFEAT]
        const float* __restrict__ bias, // [HFEAT]
        float* __restrict__ out) {      // [NNODES, HFEAT]
    __shared__ float As[16 * TFEAT];                // 8 KB A tile

    const int tid  = threadIdx.x;
    const int lane = tid & 31;
    const int wave = tid >> 5;                      // column tile 0..7
    const int row0 = blockIdx.x << 4;
    const int col0 = wave << 4;
    const int lmod = lane & 15;
    const int koff = (lane >= 16) ? 2 : 0;          // K split across lane halves

    // cooperative A-tile load: 2048 floats = 512 float4, 2 per thread
    {
        const float4* src = (const float4*)(X + (size_t)row0 * TFEAT);
        float4* dst = (float4*)As;
        dst[tid]       = src[tid];
        dst[tid + 256] = src[tid + 256];
    }
    __syncthreads();

    // accumulator preloaded with bias (same value in all 8 M-rows of a lane)
    float bv = bias[col0 + lmod];
    v8f acc;
#pragma unroll
    for (int j = 0; j < 8; ++j) acc[j] = bv;

    // A frag from LDS: lane -> row lmod, K = k+koff .. k+koff+1 (contiguous)
    const float* arow = As + lmod * TFEAT + koff;
    // B frag from global: lane -> col (col0+lmod), K stride = HFEAT
    const float* bcol = W + (size_t)koff * HFEAT + (col0 + lmod);

#pragma unroll 8
    for (int k = 0; k < TFEAT; k += 4) {
        v2f a, b;
        a.x = arow[k];
        a.y = arow[k + 1];
        b.x = bcol[(size_t)k * HFEAT];
        b.y = bcol[(size_t)(k + 1) * HFEAT];
        acc = __builtin_amdgcn_wmma_f32_16x16x4_f32(
                /*neg_a=*/false, a, /*neg_b=*/false, b,
                /*c_mod=*/(short)0, acc, /*reuse_a=*/false, /*reuse_b=*/false);
    }

    // D layout: VGPR j -> M = row0 + j (+8 for hi lanes), N = col0 + lmod
    float* orow = out + (size_t)(row0 + ((lane >= 16) ? 8 : 0)) * HFEAT
                      + (col0 + lmod);
#pragma unroll
    for (int j = 0; j < 8; ++j) orow[(size_t)j * HFEAT] = acc[j];
}

// ---------------------------------------------------------------------------

static inline size_t align_up(size_t v, size_t a) { return (v + a - 1) & ~(a - 1); }

extern "C" void kernel_launch(void* const* d_in, const int* in_sizes, int n_in,
                              void* d_out, int out_size, void* d_ws, size_t ws_size,
                              hipStream_t stream) {
    const float* x   = (const float*)d_in[0];          // [N, T]
    const int*   ei  = (const int*)d_in[1];            // [2, E]
    const float* ew  = (const float*)d_in[2];          // [E]
    const float* W   = (const float*)d_in[3];          // [T, H]
    const float* b   = (const float*)d_in[4];          // [H]
    float* out = (float*)d_out;                        // [N, H]

    const int n = NNODES, e = NEDGES;
    const int* row = ei;                                // edge_index[0]
    const int* col = ei + e;                            // edge_index[1]

    // workspace carve-up
    char* ws = (char*)d_ws;
    float* dinv = (float*)ws;                           // N floats (deg->dinv)
    ws += align_up((size_t)n * sizeof(float), 256);
    float* norm = (float*)ws;                           // E floats
    ws += align_up((size_t)e * sizeof(float), 256);
    float* buf1 = (float*)ws;                           // N*T floats
    ws += align_up((size_t)n * TFEAT * sizeof(float), 256);
    float* buf2 = (float*)ws;                           // N*T floats
    (void)ws_size; (void)n_in; (void)in_sizes; (void)out_size;

    const int TB = 256;

    // 1) degree + dinv
    init_deg_kernel<<<(n + TB - 1) / TB, TB, 0, stream>>>(dinv, n);
    accum_deg_kernel<<<(e + TB - 1) / TB, TB, 0, stream>>>(ew, col, dinv, e);
    dinv_kernel<<<(n + TB - 1) / TB, TB, 0, stream>>>(dinv, n);

    // 2) per-edge norm
    norm_kernel<<<(e + TB - 1) / TB, TB, 0, stream>>>(ew, row, col, dinv, norm, e);

    // 3) two propagation hops (x -> buf1 -> buf2)
    const int initThreads = n * 32;                      // one float4 per thread
    const int edgeBlocks  = e / EDGES_PER_BLOCK;         // exact: 200000

    prop_init_kernel<<<(initThreads + TB - 1) / TB, TB, 0, stream>>>(x, dinv, buf1, n);
    prop_edges_kernel<<<edgeBlocks, TB, 0, stream>>>(x, row, col, norm, buf1);

    prop_init_kernel<<<(initThreads + TB - 1) / TB, TB, 0, stream>>>(buf1, dinv, buf2, n);
    prop_edges_kernel<<<edgeBlocks, TB, 0, stream>>>(buf1, row, col, norm, buf2);

    // 4) out = buf2 @ W + b  (fp32 WMMA, 3125 row-tiles x 8 col-tiles)
    gemm_wmma_f32_kernel<<<NNODES / 16, 256, 0, stream>>>(buf2, W, b, out);
}